// SinkhornIteration_4234837754210
// MI455X (gfx1250) — compile-verified
//
#include <hip/hip_runtime.h>
#include <cmath>

typedef float    v2f __attribute__((ext_vector_type(2)));
typedef float    v8f __attribute__((ext_vector_type(8)));
typedef _Float16 v8h __attribute__((ext_vector_type(8)));

#define LAM      10.0f
#define INV_LAM  0.1f
#define MAX_ITER 50
#define KD       256
#define TPB      256
#define NSPLIT   32     // row-splits for the column-LSE pass

// ---------------------------------------------------------------------------
// 8-wide loaders: fp16 path is one global_load_b128; f32 path is scalar b32
// (f32 C lives at d_out+1, which is only 4B-aligned).
// ---------------------------------------------------------------------------
__device__ __forceinline__ void load8(const _Float16* p, float (&o)[8]) {
  v8h t = *(const v8h*)p;
  #pragma unroll
  for (int u = 0; u < 8; ++u) o[u] = (float)t[u];
}
__device__ __forceinline__ void load8(const float* p, float (&o)[8]) {
  #pragma unroll
  for (int u = 0; u < 8; ++u) o[u] = p[u];
}

// ---------------------------------------------------------------------------
// Row squared-norms of X and Y; also zero-initializes g (needed by iter 1).
// ---------------------------------------------------------------------------
__global__ __launch_bounds__(256) void norms_kernel(const float* __restrict__ X,
                                                    const float* __restrict__ Y,
                                                    float* __restrict__ x2,
                                                    float* __restrict__ y2,
                                                    float* __restrict__ g,
                                                    int N) {
  __shared__ float sm[TPB];
  const int b = blockIdx.x;
  const float* src = (b < N) ? (X + (size_t)b * KD) : (Y + (size_t)(b - N) * KD);
  float v = src[threadIdx.x];
  sm[threadIdx.x] = v * v;
  __syncthreads();
  for (int off = TPB / 2; off > 0; off >>= 1) {
    if (threadIdx.x < off) sm[threadIdx.x] += sm[threadIdx.x + off];
    __syncthreads();
  }
  if (threadIdx.x == 0) {
    if (b < N) x2[b] = sm[0];
    else { y2[b - N] = sm[0]; g[b - N] = 0.0f; }
  }
}

// ---------------------------------------------------------------------------
// C = sqrt(max(x2 + y2 - 2*X@Y^T, 0) + 1e-6) via V_WMMA_F32_16X16X4_F32.
// 8 waves/block arranged 4(M)x2(N); each wave owns a 32x32 tile (2x2 WMMA
// accumulators). Writes f32 C (non-temporal; only re-read once at the end)
// and, when available, an fp16 shadow copy Ch (regular stores -> L2-resident,
// 128 MB fits the 192 MB L2) used by the 100 iteration passes.
// ---------------------------------------------------------------------------
__global__ __launch_bounds__(256) void cost_gemm_kernel(const float* __restrict__ X,
                                                        const float* __restrict__ Y,
                                                        const float* __restrict__ x2,
                                                        const float* __restrict__ y2,
                                                        float* __restrict__ C,
                                                        _Float16* __restrict__ Ch,
                                                        int N, int M) {
  const int lane = threadIdx.x & 31;
  const int wid  = threadIdx.x >> 5;   // 0..7
  const int wm   = wid & 3;            // 4 waves along rows
  const int wn   = wid >> 2;           // 2 waves along cols
  const int llo  = lane & 15;
  const int lhi  = lane >> 4;

  const int m0 = blockIdx.x * 128 + wm * 32;
  const int n0 = blockIdx.y * 64  + wn * 32;

  const float* A0 = X + (size_t)(m0 + llo) * KD;
  const float* A1 = X + (size_t)(m0 + 16 + llo) * KD;
  const float* B0 = Y + (size_t)(n0 + llo) * KD;
  const float* B1 = Y + (size_t)(n0 + 16 + llo) * KD;
  const int ko = 2 * lhi;

  v8f acc00 = {}; v8f acc01 = {}; v8f acc10 = {}; v8f acc11 = {};

  #pragma unroll 4
  for (int kb = 0; kb < KD; kb += 4) {
    v2f a0 = *(const v2f*)(A0 + kb + ko);
    v2f a1 = *(const v2f*)(A1 + kb + ko);
    v2f b0 = *(const v2f*)(B0 + kb + ko);
    v2f b1 = *(const v2f*)(B1 + kb + ko);
    acc00 = __builtin_amdgcn_wmma_f32_16x16x4_f32(false, a0, false, b0, (short)0, acc00, false, false);
    acc01 = __builtin_amdgcn_wmma_f32_16x16x4_f32(false, a0, false, b1, (short)0, acc01, false, false);
    acc10 = __builtin_amdgcn_wmma_f32_16x16x4_f32(false, a1, false, b0, (short)0, acc10, false, false);
    acc11 = __builtin_amdgcn_wmma_f32_16x16x4_f32(false, a1, false, b1, (short)0, acc11, false, false);
  }

  v8f accs[2][2] = { {acc00, acc01}, {acc10, acc11} };
  #pragma unroll
  for (int tm = 0; tm < 2; ++tm) {
    #pragma unroll
    for (int tn = 0; tn < 2; ++tn) {
      v8f a = accs[tm][tn];
      const int rbase = m0 + tm * 16 + 8 * lhi;
      const int c     = n0 + tn * 16 + llo;
      const float yc = y2[c];
      #pragma unroll
      for (int v = 0; v < 8; ++v) {
        const int r = rbase + v;
        float sq = x2[r] + yc - 2.0f * a[v];
        float cv = sqrtf(fmaxf(sq, 0.0f) + 1e-6f);
        __builtin_nontemporal_store(cv, &C[(size_t)r * M + c]);
        if (Ch) Ch[(size_t)r * M + c] = (_Float16)cv;
      }
    }
  }
}

// ---------------------------------------------------------------------------
// f[row] = LAM*log(1/n) - LAM * logsumexp_j((g[j] - C[row,j]) / LAM)
// One block per row. Each thread: 8 consecutive elements per step (b128 on
// fp16 path), 8 independent branchless online-(m,s) pairs for ILP, merged
// then LDS-tree reduced.
// ---------------------------------------------------------------------------
template <typename T>
__global__ __launch_bounds__(256) void row_lse_kernel(const T* __restrict__ C,
                                                      const float* __restrict__ g,
                                                      float* __restrict__ f,
                                                      int M, float la) {
  __shared__ float sm[TPB], ss[TPB];
  const int row = blockIdx.x;
  const T* Crow = C + (size_t)row * M;

  float am[8], as[8];
  #pragma unroll
  for (int u = 0; u < 8; ++u) { am[u] = -3.0e38f; as[u] = 0.0f; }

  for (int j0 = threadIdx.x * 8; j0 < M; j0 += TPB * 8) {
    float c[8], gv[8];
    load8(Crow + j0, c);
    load8(g + j0, gv);
    #pragma unroll
    for (int u = 0; u < 8; ++u) {
      float v  = (gv[u] - c[u]) * INV_LAM;
      float mn = fmaxf(am[u], v);
      as[u] = as[u] * __expf(am[u] - mn) + __expf(v - mn);
      am[u] = mn;
    }
  }
  float m = am[0], s = as[0];
  #pragma unroll
  for (int u = 1; u < 8; ++u) {
    float mm = fmaxf(m, am[u]);
    s = s * __expf(m - mm) + as[u] * __expf(am[u] - mm);
    m = mm;
  }

  sm[threadIdx.x] = m; ss[threadIdx.x] = s;
  __syncthreads();
  for (int off = TPB / 2; off > 0; off >>= 1) {
    if (threadIdx.x < off) {
      float m1 = sm[threadIdx.x],       s1 = ss[threadIdx.x];
      float m2 = sm[threadIdx.x + off], s2 = ss[threadIdx.x + off];
      float mm = fmaxf(m1, m2);
      sm[threadIdx.x] = mm;
      ss[threadIdx.x] = s1 * __expf(m1 - mm) + s2 * __expf(m2 - mm);
    }
    __syncthreads();
  }
  if (threadIdx.x == 0)
    f[row] = LAM * la - LAM * (sm[0] + __logf(ss[0]));
}

// ---------------------------------------------------------------------------
// Column logsumexp, stage 1: each thread owns 8 consecutive columns (b128 row
// segments on fp16 path) over one of NSPLIT row-chunks. grid = (M/2048, NSPLIT).
// ---------------------------------------------------------------------------
template <typename T>
__global__ __launch_bounds__(256) void col_lse_partial_kernel(const T* __restrict__ C,
                                                              const float* __restrict__ f,
                                                              float* __restrict__ pm,
                                                              float* __restrict__ ps,
                                                              int N, int M) {
  const int col0  = (blockIdx.x * TPB + threadIdx.x) * 8;
  const int split = blockIdx.y;
  const int rows  = N / NSPLIT;
  const int r0    = split * rows;
  const T* Cp = C + (size_t)r0 * M + col0;

  float am[8], as[8];
  #pragma unroll
  for (int u = 0; u < 8; ++u) { am[u] = -3.0e38f; as[u] = 0.0f; }

  for (int i = 0; i < rows; ++i) {
    const float fv = f[r0 + i];
    float c[8];
    load8(Cp + (size_t)i * M, c);
    #pragma unroll
    for (int u = 0; u < 8; ++u) {
      float v  = (fv - c[u]) * INV_LAM;
      float mn = fmaxf(am[u], v);
      as[u] = as[u] * __expf(am[u] - mn) + __expf(v - mn);
      am[u] = mn;
    }
  }
  #pragma unroll
  for (int u = 0; u < 8; ++u) {
    pm[(size_t)split * M + col0 + u] = am[u];
    ps[(size_t)split * M + col0 + u] = as[u];
  }
}

// Stage 2: merge NSPLIT partials per column, finish g update.
__global__ __launch_bounds__(256) void col_lse_combine_kernel(const float* __restrict__ pm,
                                                              const float* __restrict__ ps,
                                                              float* __restrict__ g,
                                                              int M, float lb) {
  const int col = blockIdx.x * TPB + threadIdx.x;
  float m = -3.0e38f, s = 0.0f;
  for (int k = 0; k < NSPLIT; ++k) {
    float m2 = pm[(size_t)k * M + col];
    float s2 = ps[(size_t)k * M + col];
    float mm = fmaxf(m, m2);
    s = s * __expf(m - mm) + s2 * __expf(m2 - mm);
    m = mm;
  }
  g[col] = LAM * lb - LAM * (m + __logf(s));
}

// ---------------------------------------------------------------------------
// pi = exp((f + g^T - C)/LAM) written IN PLACE over f32 C (C lives at d_out+1).
// Last use of C and pi is never re-read -> non-temporal load + store so this
// 512 MB of traffic does not evict anything useful from L2.
// ---------------------------------------------------------------------------
__global__ __launch_bounds__(256) void pi_cost_kernel(float* __restrict__ C,
                                                      const float* __restrict__ f,
                                                      const float* __restrict__ g,
                                                      float* __restrict__ partial,
                                                      int M) {
  __shared__ float sm[TPB];
  const int row = blockIdx.x;
  const float fr = f[row];
  float* Crow = C + (size_t)row * M;
  float acc = 0.0f;
  for (int j = threadIdx.x; j < M; j += TPB) {
    float c = __builtin_nontemporal_load(&Crow[j]);
    float p = __expf((fr + g[j] - c) * INV_LAM);
    __builtin_nontemporal_store(p, &Crow[j]);
    acc += p * c;
  }
  sm[threadIdx.x] = acc;
  __syncthreads();
  for (int off = TPB / 2; off > 0; off >>= 1) {
    if (threadIdx.x < off) sm[threadIdx.x] += sm[threadIdx.x + off];
    __syncthreads();
  }
  if (threadIdx.x == 0) partial[row] = sm[0];
}

// Deterministic final reduction (no float atomics): fixed-order tree.
__global__ __launch_bounds__(256) void final_reduce_kernel(const float* __restrict__ partial,
                                                           int n, float* __restrict__ out) {
  __shared__ float sm[TPB];
  float a = 0.0f;
  for (int i = threadIdx.x; i < n; i += TPB) a += partial[i];
  sm[threadIdx.x] = a;
  __syncthreads();
  for (int off = TPB / 2; off > 0; off >>= 1) {
    if (threadIdx.x < off) sm[threadIdx.x] += sm[threadIdx.x + off];
    __syncthreads();
  }
  if (threadIdx.x == 0) out[0] = sm[0];
}

extern "C" void kernel_launch(void* const* d_in, const int* in_sizes, int n_in,
                              void* d_out, int out_size, void* d_ws, size_t ws_size,
                              hipStream_t stream) {
  (void)n_in; (void)out_size;
  const float* X = (const float*)d_in[0];
  const float* Y = (const float*)d_in[1];
  float* out = (float*)d_out;
  float* ws  = (float*)d_ws;

  const int N = in_sizes[0] / KD;   // 8192
  const int M = in_sizes[1] / KD;   // 8192

  // d_ws layout (floats), then 256B-aligned fp16 shadow cost matrix.
  float* x2      = ws;
  float* y2      = x2 + N;
  float* f       = y2 + M;
  float* g       = f + N;
  float* pm      = g + M;
  float* ps      = pm + (size_t)NSPLIT * M;
  float* partial = ps + (size_t)NSPLIT * M;
  size_t small_bytes = (size_t)((partial + N) - ws) * sizeof(float);
  size_t ch_off      = (small_bytes + 255) & ~(size_t)255;
  bool   use_half    = ws_size >= ch_off + (size_t)N * (size_t)M * sizeof(_Float16);
  _Float16* Ch = use_half ? (_Float16*)((char*)d_ws + ch_off) : nullptr;

  // f32 cost matrix lives where pi will end up; pi_cost_kernel overwrites in place.
  float* C = out + 1;

  const float la = -logf((float)N);  // log(1/n)
  const float lb = -logf((float)M);  // log(1/m)

  norms_kernel<<<N + M, TPB, 0, stream>>>(X, Y, x2, y2, g, N);
  cost_gemm_kernel<<<dim3(N / 128, M / 64), TPB, 0, stream>>>(X, Y, x2, y2, C, Ch, N, M);

  for (int it = 0; it < MAX_ITER; ++it) {
    if (use_half) {
      row_lse_kernel<_Float16><<<N, TPB, 0, stream>>>(Ch, g, f, M, la);
      col_lse_partial_kernel<_Float16><<<dim3(M / (TPB * 8), NSPLIT), TPB, 0, stream>>>(Ch, f, pm, ps, N, M);
    } else {
      row_lse_kernel<float><<<N, TPB, 0, stream>>>(C, g, f, M, la);
      col_lse_partial_kernel<float><<<dim3(M / (TPB * 8), NSPLIT), TPB, 0, stream>>>(C, f, pm, ps, N, M);
    }
    col_lse_combine_kernel<<<M / TPB, TPB, 0, stream>>>(pm, ps, g, M, lb);
  }

  pi_cost_kernel<<<N, TPB, 0, stream>>>(C, f, g, partial, M);
  final_reduce_kernel<<<1, TPB, 0, stream>>>(partial, N, out);
}